// LlamaCatFlow_1219770712332
// MI455X (gfx1250) — compile-verified
//
#include <hip/hip_runtime.h>

typedef float v2f __attribute__((ext_vector_type(2)));
typedef float v8f __attribute__((ext_vector_type(8)));

#define TEMP_INV 1.25f     // 1/0.8
#define FM_EPS   1e-10f
#define DDIM     16
#define MAXV     8192      // reference V; snrm LDS sized for this
#define NEGINF  -3.402823466e38f
#define WPT      4                       // waves cooperating per 16-row tile
#define TPB      256
#define NWAVES   (TPB / 32)
#define ROWS_PER_BLOCK ((NWAVES / WPT) * 16)

// lane <-> lane^16 exchange on the VALU pipe (v_permlanex16_b32, identity
// cross-selectors) instead of ds_bpermute.
__device__ __forceinline__ float xor16f(float x) {
  const int v = __builtin_amdgcn_permlanex16(
      __float_as_int(x), __float_as_int(x), 0x76543210, 0xfedcba98, false, false);
  return __int_as_float(v);
}
__device__ __forceinline__ int xor16i(int x) {
  return __builtin_amdgcn_permlanex16(x, x, 0x76543210, 0xfedcba98, false, false);
}

// T-layout (lane: row=col fixed, elem i -> dim i+8*hi) -> B-fragment layout
// (lane: N=col, K pairs {4kc+2hi, 4kc+2hi+1}) for V_WMMA_F32_16X16X4_F32.
__device__ __forceinline__ void t2b(const float xT[8], float xB[8], int hi) {
  float q[8];
#pragma unroll
  for (int i = 0; i < 8; ++i) q[i] = xor16f(xT[i]);
  xB[0] = hi ? q[2] : xT[0];
  xB[1] = hi ? q[3] : xT[1];
  xB[2] = hi ? q[6] : xT[4];
  xB[3] = hi ? q[7] : xT[5];
  xB[4] = hi ? xT[2] : q[0];
  xB[5] = hi ? xT[3] : q[1];
  xB[6] = hi ? xT[6] : q[4];
  xB[7] = hi ? xT[7] : q[5];
}

// One 16-wide chunk of transposed logits: logitsT[v][row] = W^T x + b + t*tp,
// bias injected as the WMMA C initializer. C elem i -> v = vb+i+8hi, row = col.
__device__ __forceinline__ v8f logits_chunk(
    const float* __restrict__ W, const float* __restrict__ bias,
    const float* __restrict__ tproj, int V, int vb, float t,
    const float xB[8], int col, int hi) {
  float wa[8];                               // A-frag: M=v(=vb+col), K=d
#pragma unroll
  for (int kc = 0; kc < 4; ++kc) {
    const int d0 = 4 * kc + 2 * hi;
    wa[2 * kc + 0] = W[d0 * V + vb + col];
    wa[2 * kc + 1] = W[(d0 + 1) * V + vb + col];
  }
  const float4 b0 = *(const float4*)&bias[vb + 8 * hi];
  const float4 b1 = *(const float4*)&bias[vb + 8 * hi + 4];
  const float4 t0 = *(const float4*)&tproj[vb + 8 * hi];
  const float4 t1 = *(const float4*)&tproj[vb + 8 * hi + 4];
  v8f acc = {b0.x + t * t0.x, b0.y + t * t0.y, b0.z + t * t0.z, b0.w + t * t0.w,
             b1.x + t * t1.x, b1.y + t * t1.y, b1.z + t * t1.z, b1.w + t * t1.w};
#pragma unroll
  for (int kc = 0; kc < 4; ++kc) {
    v2f a = {wa[2 * kc], wa[2 * kc + 1]};
    v2f b = {xB[2 * kc], xB[2 * kc + 1]};
    acc = __builtin_amdgcn_wmma_f32_16x16x4_f32(false, a, false, b,
                                                (short)0, acc, false, false);
  }
  return acc;
}

// velocity(t,x) for one 16-row tile, V-range [vlo,vhi) per wave; the 4 waves
// of a tile group combine rowmax / sum-exp / mu partials through LDS.
// vlo/vhi/wave/g0 are readfirstlane-uniform => scalar loop control.
__device__ __forceinline__ void velocity(
    const float* __restrict__ W, const float* __restrict__ bias,
    const float* __restrict__ tproj, const float* __restrict__ cb,
    int V, int vlo, int vhi, float t,
    const float xB[8], const float xT[8], float vT[8],
    int col, int hi, int wave, int lane, int g0,
    float (*rmw)[32], float (*psw)[32], float (*maccw)[8][32]) {
  // ---- pass A: partial per-row max of (unscaled) logits ----
  float rm = NEGINF;
  for (int vb = vlo; vb < vhi; vb += 16) {
    const v8f acc = logits_chunk(W, bias, tproj, V, vb, t, xB, col, hi);
#pragma unroll
    for (int i = 0; i < 8; ++i) rm = fmaxf(rm, acc[i]);
  }
  rm = fmaxf(rm, xor16f(rm));                // combine the two v-halves
  rmw[wave][lane] = rm;
  __syncthreads();
#pragma unroll
  for (int k = 0; k < WPT; ++k) rm = fmaxf(rm, rmw[g0 + k][lane]);

  // ---- pass B: partial exp-sum and probs^T @ codebook via WMMA ----
  float ps = 0.0f;
  v8f macc = {};
  for (int vb = vlo; vb < vhi; vb += 16) {
    const v8f acc = logits_chunk(W, bias, tproj, V, vb, t, xB, col, hi);
    float p[8];
#pragma unroll
    for (int i = 0; i < 8; ++i) {
      p[i] = __expf((acc[i] - rm) * TEMP_INV);
      ps += p[i];
    }
    float pb[8];                             // probsT C-layout -> B-frag
    t2b(p, pb, hi);
    float ca[8];                             // A-frag of cb^T: M=d(=col), K=v
#pragma unroll
    for (int kc = 0; kc < 4; ++kc) {
      const int v0 = vb + 4 * kc + 2 * hi;
      ca[2 * kc + 0] = cb[v0 * DDIM + col];
      ca[2 * kc + 1] = cb[(v0 + 1) * DDIM + col];
    }
#pragma unroll
    for (int kc = 0; kc < 4; ++kc) {
      v2f a = {ca[2 * kc], ca[2 * kc + 1]};
      v2f b = {pb[2 * kc], pb[2 * kc + 1]};
      macc = __builtin_amdgcn_wmma_f32_16x16x4_f32(false, a, false, b,
                                                   (short)0, macc, false, false);
    }
  }
  ps += xor16f(ps);
  psw[wave][lane] = ps;
#pragma unroll
  for (int i = 0; i < 8; ++i) maccw[wave][i][lane] = macc[i];
  __syncthreads();
  float pst = 0.0f;
  v8f msum = {};
#pragma unroll
  for (int k = 0; k < WPT; ++k) {
    pst += psw[g0 + k][lane];
#pragma unroll
    for (int i = 0; i < 8; ++i) msum[i] += maccw[g0 + k][i][lane];
  }
  const float rps = 1.0f / pst;
  const float sc = 1.0f / (1.0f - t + FM_EPS);
#pragma unroll
  for (int i = 0; i < 8; ++i)
    vT[i] = (msum[i] * rps - xT[i]) * sc;    // (mu - x)/(1-t+eps), T layout
}

__global__ void __launch_bounds__(TPB)
catflow_kernel(const float* __restrict__ x0, const float* __restrict__ cb,
               const float* __restrict__ W, const float* __restrict__ bias,
               const float* __restrict__ tproj, const int* __restrict__ nstep_p,
               float* __restrict__ out, int rows, int V) {
  __shared__ float snrm[MAXV];               // ||c_v||^2 table
  __shared__ float rmw[NWAVES][32];          // pass-A partials
  __shared__ float psw[NWAVES][32];          // pass-B partials
  __shared__ float maccw[NWAVES][8][32];
  __shared__ float bestw[NWAVES][32];        // argmin partials
  __shared__ int   bidxw[NWAVES][32];

  // Cooperative codebook-norm precompute (before any early-out => barrier safe).
  for (int v = threadIdx.x; v < V; v += TPB) {
    const float4 c0 = *(const float4*)&cb[v * DDIM + 0];
    const float4 c1 = *(const float4*)&cb[v * DDIM + 4];
    const float4 c2 = *(const float4*)&cb[v * DDIM + 8];
    const float4 c3 = *(const float4*)&cb[v * DDIM + 12];
    snrm[v] = c0.x * c0.x + c0.y * c0.y + c0.z * c0.z + c0.w * c0.w +
              c1.x * c1.x + c1.y * c1.y + c1.z * c1.z + c1.w * c1.w +
              c2.x * c2.x + c2.y * c2.y + c2.z * c2.z + c2.w * c2.w +
              c3.x * c3.x + c3.y * c3.y + c3.z * c3.z + c3.w * c3.w;
  }
  __syncthreads();

  // Wave-uniform values forced into SGPRs => scalar loop bounds and branches.
  const int wave = __builtin_amdgcn_readfirstlane(threadIdx.x >> 5);
  const int lane = threadIdx.x & 31;
  const int col  = lane & 15;
  const int hi   = lane >> 4;
  const int tile = wave / WPT;               // tile within block
  const int wq   = wave % WPT;               // wave index within tile group
  const int g0   = wave - wq;                // first wave of group
  const int rowbase = blockIdx.x * ROWS_PER_BLOCK + tile * 16;
  if (rowbase >= rows) return;               // reference rows%32==0: no exits
  const int myrow = rowbase + col;

  const int nsteps = *nstep_p;
  const float dt = 1.0f / (float)(nsteps - 1);
  const int vchunk = (((V + WPT - 1) / WPT) + 15) & ~15;
  const int vlo = (wq * vchunk < V) ? wq * vchunk : V;
  const int vhi = (vlo + vchunk < V) ? vlo + vchunk : V;

  // x in T layout: xT[i] = x[row=myrow][d = i + 8*hi]  (two b128 loads).
  float xT[8];
  {
    const float4 a = *(const float4*)&x0[myrow * DDIM + 8 * hi];
    const float4 b = *(const float4*)&x0[myrow * DDIM + 8 * hi + 4];
    xT[0] = a.x; xT[1] = a.y; xT[2] = a.z; xT[3] = a.w;
    xT[4] = b.x; xT[5] = b.y; xT[6] = b.z; xT[7] = b.w;
  }
  float xB[8];
  t2b(xT, xB, hi);

  // Midpoint integrator: x <- x + dt * v(t + dt/2, x + dt/2 * v(t, x))
  for (int s = 0; s < nsteps - 1; ++s) {
    const float t = s * dt;
    float v1[8];
    velocity(W, bias, tproj, cb, V, vlo, vhi, t, xB, xT, v1,
             col, hi, wave, lane, g0, rmw, psw, maccw);
    float xmT[8], xmB[8];
#pragma unroll
    for (int i = 0; i < 8; ++i) xmT[i] = xT[i] + 0.5f * dt * v1[i];
    t2b(xmT, xmB, hi);
    float v2[8];
    velocity(W, bias, tproj, cb, V, vlo, vhi, t + 0.5f * dt, xmB, xmT, v2,
             col, hi, wave, lane, g0, rmw, psw, maccw);
#pragma unroll
    for (int i = 0; i < 8; ++i) xT[i] += dt * v2[i];
    t2b(xT, xB, hi);
  }

  // Store x_final (all 4 waves of a group hold identical xT; one writes).
  if (wq == 0) {
    float4 a = {xT[0], xT[1], xT[2], xT[3]};
    float4 b = {xT[4], xT[5], xT[6], xT[7]};
    *(float4*)&out[myrow * DDIM + 8 * hi]     = a;
    *(float4*)&out[myrow * DDIM + 8 * hi + 4] = b;
  }

  // VQ argmin: argmax_v ( x . c_v - 0.5*||c_v||^2 ), partial WMMA sweep.
  float best = NEGINF;
  int bi = 0;
  for (int vb = vlo; vb < vhi; vb += 16) {
    float ca[8];                             // A-frag: M=v(=vb+col), K=d
#pragma unroll
    for (int kc = 0; kc < 4; ++kc) {
      const v2f c2v = *(const v2f*)&cb[(vb + col) * DDIM + 4 * kc + 2 * hi];
      ca[2 * kc + 0] = c2v.x;
      ca[2 * kc + 1] = c2v.y;
    }
    v8f acc = {};
#pragma unroll
    for (int kc = 0; kc < 4; ++kc) {
      v2f a = {ca[2 * kc], ca[2 * kc + 1]};
      v2f b = {xB[2 * kc], xB[2 * kc + 1]};
      acc = __builtin_amdgcn_wmma_f32_16x16x4_f32(false, a, false, b,
                                                  (short)0, acc, false, false);
    }
    const float4 n0 = *(const float4*)&snrm[vb + 8 * hi];
    const float4 n1 = *(const float4*)&snrm[vb + 8 * hi + 4];
    const float nn[8] = {n0.x, n0.y, n0.z, n0.w, n1.x, n1.y, n1.z, n1.w};
#pragma unroll
    for (int i = 0; i < 8; ++i) {            // elem i -> v = vb + i + 8*hi
      const float sv = acc[i] - 0.5f * nn[i];
      if (sv > best) { best = sv; bi = vb + i + 8 * hi; }
    }
  }
  {                                          // combine the two v-halves
    const float ob = xor16f(best);
    const int   oi = xor16i(bi);
    if (ob > best || (ob == best && oi < bi)) { best = ob; bi = oi; }
  }
  bestw[wave][lane] = best;
  bidxw[wave][lane] = bi;
  __syncthreads();
  best = NEGINF; bi = 0;
#pragma unroll
  for (int k = 0; k < WPT; ++k) {            // combine the 4 V-quarters
    const float ob = bestw[g0 + k][lane];
    const int   oi = bidxw[g0 + k][lane];
    if (ob > best || (ob == best && oi < bi)) { best = ob; bi = oi; }
  }
  if (wq == 0 && hi == 0) out[rows * DDIM + myrow] = (float)bi;
}

extern "C" void kernel_launch(void* const* d_in, const int* in_sizes, int n_in,
                              void* d_out, int out_size, void* d_ws, size_t ws_size,
                              hipStream_t stream) {
  (void)n_in; (void)out_size; (void)d_ws; (void)ws_size;
  const float* x0    = (const float*)d_in[0];
  const float* cbp   = (const float*)d_in[1];
  const float* Wm    = (const float*)d_in[2];
  const float* bm    = (const float*)d_in[3];
  const float* tp    = (const float*)d_in[4];
  const int*   nstep = (const int*)d_in[5];
  float* out = (float*)d_out;

  const int rows = in_sizes[0] / DDIM;   // B*S = 8192
  const int V    = in_sizes[3];          // 8192

  dim3 block(TPB);                       // 8 wave32: 2 tiles x 4 coop waves
  dim3 grid((rows + ROWS_PER_BLOCK - 1) / ROWS_PER_BLOCK);   // 256 blocks
  catflow_kernel<<<grid, block, 0, stream>>>(x0, cbp, Wm, bm, tp, nstep, out,
                                             rows, V);
}